// Attention_37495064494240
// MI455X (gfx1250) — compile-verified
//
#include <hip/hip_runtime.h>
#include <cstdint>

// Problem constants (match reference)
#define B_   16
#define S_   626
#define H_   768
#define NH_  12
#define DH_  64
#define SP_  640          // S padded to 40 tiles of 16
#define ST_  40           // S tiles (16-row)
#define ST2_ 20           // S tiles (32-row)
#define SPP_ 648          // LDS row stride (pad to reduce bank conflicts)
#define SCALE_ 0.125f     // 1/sqrt(64)
#define CMAX_  0.25f      // COEFF_MAX

typedef float v2f __attribute__((ext_vector_type(2)));
typedef float v8f __attribute__((ext_vector_type(8)));

// D = A(16x4, f32) * B(4x16, f32) + C(16x16, f32)   -> v_wmma_f32_16x16x4_f32
static __device__ __forceinline__ v8f wmma4(v2f a, v2f b, v8f c) {
  return __builtin_amdgcn_wmma_f32_16x16x4_f32(
      /*neg_a=*/false, a, /*neg_b=*/false, b,
      /*c_mod=*/(short)0, c, /*reuse_a=*/false, /*reuse_b=*/false);
}

// 6-operand load group for the 32x64 GEMM tile (2 A rows-tiles, 4 B col-tiles)
static __device__ __forceinline__ void ld_ab6(
    const float* xr0, const float* xr1, const float* wp0, const float* wp1,
    const float* wp2, const float* wp3, int off,
    v2f& a0, v2f& a1, v2f& b0, v2f& b1, v2f& b2, v2f& b3)
{
  a0 = *(const v2f*)(xr0 + off); a1 = *(const v2f*)(xr1 + off);
  b0 = *(const v2f*)(wp0 + off); b1 = *(const v2f*)(wp1 + off);
  b2 = *(const v2f*)(wp2 + off); b3 = *(const v2f*)(wp3 + off);
}

// 8 independent WMMAs for one k-step of the 32x64 tile
static __device__ __forceinline__ void step8(
    v8f acc[2][4], v2f a0, v2f a1, v2f b0, v2f b1, v2f b2, v2f b3)
{
  acc[0][0] = wmma4(a0, b0, acc[0][0]);
  acc[0][1] = wmma4(a0, b1, acc[0][1]);
  acc[0][2] = wmma4(a0, b2, acc[0][2]);
  acc[0][3] = wmma4(a0, b3, acc[0][3]);
  acc[1][0] = wmma4(a1, b0, acc[1][0]);
  acc[1][1] = wmma4(a1, b1, acc[1][1]);
  acc[1][2] = wmma4(a1, b2, acc[1][2]);
  acc[1][3] = wmma4(a1, b3, acc[1][3]);
}

// ---------------------------------------------------------------------------
// Kernel 0: transpose the 768x768 weight matrices once, so the WMMA B-operand
// (needs (k,col),(k+1,col) per lane) becomes a contiguous b64 load.
// grid: (48, 48, 4), block: 256
// ---------------------------------------------------------------------------
__global__ __launch_bounds__(256)
void transpose_w(const float* __restrict__ Wq, const float* __restrict__ Wk,
                 const float* __restrict__ Wv, const float* __restrict__ Wo,
                 float* __restrict__ Wqt, float* __restrict__ Wkt,
                 float* __restrict__ Wvt, float* __restrict__ Wot)
{
  __shared__ float t[16][17];
  const int m = blockIdx.z;
  const float* src = (m == 0) ? Wq : (m == 1) ? Wk : (m == 2) ? Wv : Wo;
  float*       dst = (m == 0) ? Wqt : (m == 1) ? Wkt : (m == 2) ? Wvt : Wot;
  const int tx = threadIdx.x & 15, ty = threadIdx.x >> 4;
  const int k0 = blockIdx.x * 16, n0 = blockIdx.y * 16;
  t[ty][tx] = src[(size_t)(k0 + ty) * H_ + n0 + tx];
  __syncthreads();
  dst[(size_t)(n0 + ty) * H_ + k0 + tx] = t[tx][ty];   // dst[n][k] = src[k][n]
}

// ---------------------------------------------------------------------------
// Kernel 1: Q/K/V = X @ W + b.  32 rows x one head (64 cols) per wave;
// ping-pong double-buffered (no rotation copies).
// Q,K stored [b][h][s_pad][d]; V stored TRANSPOSED [b][h][d][s_pad].
// grid: (ST2_, NH_, B_*3), block: 32 (one wave)
// ---------------------------------------------------------------------------
__global__ __launch_bounds__(32)
void qkv_gemm(const float* __restrict__ X,
              const float* __restrict__ Wqt, const float* __restrict__ bq,
              const float* __restrict__ Wkt, const float* __restrict__ bk,
              const float* __restrict__ Wvt, const float* __restrict__ bv,
              float* __restrict__ Q, float* __restrict__ K, float* __restrict__ V)
{
  const int st = blockIdx.x;                 // 32-row tile
  const int hh = blockIdx.y;                 // head index (64 output cols)
  const int b  = blockIdx.z / 3;
  const int m  = blockIdx.z % 3;
  const float* Wt   = (m == 0) ? Wqt : (m == 1) ? Wkt : Wvt;
  const float* bias = (m == 0) ? bq  : (m == 1) ? bk  : bv;
  float*       Dst  = (m == 0) ? Q   : (m == 1) ? K   : V;

  const int lane = threadIdx.x;
  const int hf = lane >> 4, l16 = lane & 15;
  const int hf2 = hf * 2;
  const int m0 = st * 32;
  // clamped row pointers (garbage rows are zeroed at store; rows independent)
  const int r0 = m0 + l16,      r0c = (r0 < S_) ? r0 : (S_ - 1);
  const int r1 = m0 + 16 + l16, r1c = (r1 < S_) ? r1 : (S_ - 1);
  const float* xr0 = X + ((size_t)b * S_ + r0c) * H_;
  const float* xr1 = X + ((size_t)b * S_ + r1c) * H_;
  const int n0 = hh * DH_;
  const float* wp0 = Wt + (size_t)(n0 +  0 + l16) * H_;
  const float* wp1 = Wt + (size_t)(n0 + 16 + l16) * H_;
  const float* wp2 = Wt + (size_t)(n0 + 32 + l16) * H_;
  const float* wp3 = Wt + (size_t)(n0 + 48 + l16) * H_;

  v8f acc[2][4] = {};
  v2f pa0, pa1, pb0, pb1, pb2, pb3;   // ping buffer
  v2f qa0, qa1, qb0, qb1, qb2, qb3;   // pong buffer
  ld_ab6(xr0, xr1, wp0, wp1, wp2, wp3, hf2, pa0, pa1, pb0, pb1, pb2, pb3);
  #pragma unroll 1
  for (int k = 0; k < H_ - 8; k += 8) {
    ld_ab6(xr0, xr1, wp0, wp1, wp2, wp3, k + 4 + hf2,
           qa0, qa1, qb0, qb1, qb2, qb3);
    step8(acc, pa0, pa1, pb0, pb1, pb2, pb3);
    ld_ab6(xr0, xr1, wp0, wp1, wp2, wp3, k + 8 + hf2,
           pa0, pa1, pb0, pb1, pb2, pb3);
    step8(acc, qa0, qa1, qb0, qb1, qb2, qb3);
  }
  ld_ab6(xr0, xr1, wp0, wp1, wp2, wp3, H_ - 4 + hf2,
         qa0, qa1, qb0, qb1, qb2, qb3);
  step8(acc, pa0, pa1, pb0, pb1, pb2, pb3);
  step8(acc, qa0, qa1, qb0, qb1, qb2, qb3);

  float* base = Dst + ((size_t)b * NH_ + hh) * SP_ * DH_;
  #pragma unroll
  for (int t = 0; t < 2; ++t) {
    #pragma unroll
    for (int j = 0; j < 4; ++j) {
      const int d0 = j * 16 + l16;
      const float bsv = bias[n0 + d0];
      #pragma unroll
      for (int r = 0; r < 8; ++r) {
        const int sr = m0 + t * 16 + r + hf * 8;
        const float val = (sr < S_) ? (acc[t][j][r] + bsv) : 0.0f;
        if (m == 2) base[(size_t)d0 * SP_ + sr] = val;   // V: [d][token]
        else        base[(size_t)sr * DH_ + d0] = val;   // Q,K: [token][d]
      }
    }
  }
}

// ---------------------------------------------------------------------------
// Kernel 2: per (q-tile, head, batch):
//   scores = Q Kᵀ * 0.125 (Q A-fragment held in regs for all 40 K-tiles;
//   16 B fragments batch-loaded per tile), row-0 fixup, col-0 save, softmax,
//   weights write (b64), ctx = probs @ Vᵀ-layout (ping-pong pipelined)
// grid: (ST_, NH_, B_), block: 32 (one wave), LDS = 16*648*4 = 41.5 KB
// ---------------------------------------------------------------------------
__global__ __launch_bounds__(32)
void attn(const float* __restrict__ Q, const float* __restrict__ K,
          const float* __restrict__ Vt, const float* __restrict__ mask,
          float* __restrict__ weights, float* __restrict__ col0,
          float* __restrict__ ctx)
{
  __shared__ float sc[16][SPP_];
  const int qt = blockIdx.x;
  const int h  = blockIdx.y;
  const int b  = blockIdx.z;
  const int lane = threadIdx.x;
  const int hf = lane >> 4, l16 = lane & 15;
  const int hf2 = hf * 2;

  const size_t bh = (size_t)b * NH_ + h;
  const float* Qb  = Q  + bh * SP_ * DH_;
  const float* Kb  = K  + bh * SP_ * DH_;
  const float* Vtb = Vt + bh * DH_ * SP_;   // [d][token]

  // ---- Pass 1: scores. Preload Q-tile A fragments (16 v2f, reused 40x) ----
  const int qrow = qt * 16 + l16;
  v2f afrag[16];
  {
    const float* qp = Qb + (size_t)qrow * DH_;
    #pragma unroll
    for (int d = 0; d < 16; ++d) afrag[d] = *(const v2f*)(qp + d * 4 + hf2);
  }
  #pragma unroll 1
  for (int kt = 0; kt < ST_; ++kt) {
    const float* kp = Kb + (size_t)(kt * 16 + l16) * DH_;
    __builtin_prefetch(kp + 16 * DH_, 0, 1);   // next K tile (speculative)
    v2f bfrag[16];
    #pragma unroll
    for (int d = 0; d < 16; ++d) bfrag[d] = *(const v2f*)(kp + d * 4 + hf2);
    v8f acc = {};
    #pragma unroll
    for (int d = 0; d < 16; ++d) acc = wmma4(afrag[d], bfrag[d], acc);
    #pragma unroll
    for (int r = 0; r < 8; ++r)
      sc[r + hf * 8][kt * 16 + l16] = acc[r] * SCALE_;
  }
  __syncthreads();

  // ---- row-0 special case (only the tile containing q==0) ----------------
  if (qt == 0) {
    float mx = -3.0e38f;
    for (int k = lane; k < S_; k += 32) mx = fmaxf(mx, sc[0][k]);
    #pragma unroll
    for (int off = 16; off; off >>= 1) mx = fmaxf(mx, __shfl_xor(mx, off, 32));
    for (int k = lane; k < S_; k += 32) {
      const float mk = (k == 0) ? 0.0f : mask[(size_t)b * (S_ - 1) + (k - 1)];
      if (mk < 0.5f) sc[0][k] += mx * CMAX_;
    }
  }
  __syncthreads();

  // ---- save pre-softmax column-0 scores (post row-0 fix) for sm2 ---------
  if (lane < 16)
    col0[bh * SP_ + qt * 16 + lane] = sc[lane][0];

  // ---- softmax over k for each of 16 rows (lane pair l16 / l16+16) -------
  {
    const int row = l16;
    float mx = -3.0e38f;
    for (int k = hf; k < S_; k += 2) mx = fmaxf(mx, sc[row][k]);
    mx = fmaxf(mx, __shfl_xor(mx, 16, 32));
    float sum = 0.0f;
    for (int k = hf; k < S_; k += 2) {
      const float e = __expf(sc[row][k] - mx);
      sc[row][k] = e;
      sum += e;
    }
    sum += __shfl_xor(sum, 16, 32);
    const float inv = 1.0f / sum;
    for (int k = hf; k < S_; k += 2)  sc[row][k] *= inv;
    for (int k = S_ + hf; k < SP_; k += 2) sc[row][k] = 0.0f;  // zero pad cols
  }
  __syncthreads();

  // ---- write attention weights (b64 stores; S_ even, stride 8B-aligned) --
  {
    const int qrem = S_ - qt * 16;
    const int qmax = (qrem < 16) ? qrem : 16;
    for (int r = 0; r < qmax; ++r) {
      float* wr = weights + (bh * S_ + qt * 16 + r) * S_;
      for (int k = lane * 2; k < S_; k += 64)
        *(v2f*)(wr + k) = *(const v2f*)(&sc[r][k]);
    }
  }

  // ---- ctx = probs @ V  (16 x 64; ping-pong pipelined) -------------------
  {
    const float* vp0 = Vtb + (size_t)( 0 + l16) * SP_;
    const float* vp1 = Vtb + (size_t)(16 + l16) * SP_;
    const float* vp2 = Vtb + (size_t)(32 + l16) * SP_;
    const float* vp3 = Vtb + (size_t)(48 + l16) * SP_;
    const float* ap = &sc[l16][0];
    v8f acc[4] = {};
    v2f pa, pb0, pb1, pb2, pb3;   // ping
    v2f qa, qb0, qb1, qb2, qb3;   // pong
    pa  = *(const v2f*)(ap + hf2);
    pb0 = *(const v2f*)(vp0 + hf2); pb1 = *(const v2f*)(vp1 + hf2);
    pb2 = *(const v2f*)(vp2 + hf2); pb3 = *(const v2f*)(vp3 + hf2);
    #pragma unroll 1
    for (int k = 0; k < SP_ - 8; k += 8) {
      const int k1 = k + 4 + hf2, k2 = k + 8 + hf2;
      qa  = *(const v2f*)(ap + k1);
      qb0 = *(const v2f*)(vp0 + k1); qb1 = *(const v2f*)(vp1 + k1);
      qb2 = *(const v2f*)(vp2 + k1); qb3 = *(const v2f*)(vp3 + k1);
      acc[0] = wmma4(pa, pb0, acc[0]);
      acc[1] = wmma4(pa, pb1, acc[1]);
      acc[2] = wmma4(pa, pb2, acc[2]);
      acc[3] = wmma4(pa, pb3, acc[3]);
      pa  = *(const v2f*)(ap + k2);
      pb0 = *(const v2f*)(vp0 + k2); pb1 = *(const v2f*)(vp1 + k2);
      pb2 = *(const v2f*)(vp2 + k2); pb3 = *(const v2f*)(vp3 + k2);
      acc[0] = wmma4(qa, qb0, acc[0]);
      acc[1] = wmma4(qa, qb1, acc[1]);
      acc[2] = wmma4(qa, qb2, acc[2]);
      acc[3] = wmma4(qa, qb3, acc[3]);
    }
    {
      const int k1 = SP_ - 4 + hf2;
      qa  = *(const v2f*)(ap + k1);
      qb0 = *(const v2f*)(vp0 + k1); qb1 = *(const v2f*)(vp1 + k1);
      qb2 = *(const v2f*)(vp2 + k1); qb3 = *(const v2f*)(vp3 + k1);
      acc[0] = wmma4(pa, pb0, acc[0]);
      acc[1] = wmma4(pa, pb1, acc[1]);
      acc[2] = wmma4(pa, pb2, acc[2]);
      acc[3] = wmma4(pa, pb3, acc[3]);
      acc[0] = wmma4(qa, qb0, acc[0]);
      acc[1] = wmma4(qa, qb1, acc[1]);
      acc[2] = wmma4(qa, qb2, acc[2]);
      acc[3] = wmma4(qa, qb3, acc[3]);
    }
    #pragma unroll
    for (int j = 0; j < 4; ++j) {
      #pragma unroll
      for (int r = 0; r < 8; ++r) {
        const int q = qt * 16 + r + hf * 8;
        ctx[((size_t)b * SP_ + q) * H_ + h * DH_ + j * 16 + l16] = acc[j][r];
      }
    }
  }
}

// ---------------------------------------------------------------------------
// Kernel 3: sm2 = softmax over queries of column-0 scores, per (b,h)
// grid: (B_*NH_), block: 32
// ---------------------------------------------------------------------------
__global__ __launch_bounds__(32)
void sm2_softmax(const float* __restrict__ col0, float* __restrict__ sm2)
{
  const int bh = blockIdx.x;
  const float* c = col0 + (size_t)bh * SP_;
  const int lane = threadIdx.x;

  float mx = -3.0e38f;
  for (int q = lane; q < S_; q += 32) mx = fmaxf(mx, c[q]);
  #pragma unroll
  for (int off = 16; off; off >>= 1) mx = fmaxf(mx, __shfl_xor(mx, off, 32));

  float sum = 0.0f;
  for (int q = lane; q < S_; q += 32) sum += __expf(c[q] - mx);
  #pragma unroll
  for (int off = 16; off; off >>= 1) sum += __shfl_xor(sum, off, 32);

  const float inv = 1.0f / sum;
  for (int q = lane; q < S_; q += 32)
    sm2[(size_t)bh * S_ + q] = __expf(c[q] - mx) * inv;
}

// ---------------------------------------------------------------------------
// Kernel 4: out = ctx @ Wo + bo.  32 rows x 64 cols per wave, ping-pong.
// grid: (ST2_, 12, B_), block: 32
// ---------------------------------------------------------------------------
__global__ __launch_bounds__(32)
void out_gemm(const float* __restrict__ ctx, const float* __restrict__ Wot,
              const float* __restrict__ bo, float* __restrict__ out)
{
  const int st = blockIdx.x;
  const int ct = blockIdx.y;                 // 64-col tile
  const int b  = blockIdx.z;
  const int lane = threadIdx.x;
  const int hf = lane >> 4, l16 = lane & 15;
  const int hf2 = hf * 2;
  const int m0 = st * 32;
  const float* xr0 = ctx + ((size_t)b * SP_ + m0 + l16) * H_;      // padded
  const float* xr1 = ctx + ((size_t)b * SP_ + m0 + 16 + l16) * H_;
  const int n0 = ct * 64;
  const float* wp0 = Wot + (size_t)(n0 +  0 + l16) * H_;
  const float* wp1 = Wot + (size_t)(n0 + 16 + l16) * H_;
  const float* wp2 = Wot + (size_t)(n0 + 32 + l16) * H_;
  const float* wp3 = Wot + (size_t)(n0 + 48 + l16) * H_;

  v8f acc[2][4] = {};
  v2f pa0, pa1, pb0, pb1, pb2, pb3;   // ping
  v2f qa0, qa1, qb0, qb1, qb2, qb3;   // pong
  ld_ab6(xr0, xr1, wp0, wp1, wp2, wp3, hf2, pa0, pa1, pb0, pb1, pb2, pb3);
  #pragma unroll 1
  for (int k = 0; k < H_ - 8; k += 8) {
    ld_ab6(xr0, xr1, wp0, wp1, wp2, wp3, k + 4 + hf2,
           qa0, qa1, qb0, qb1, qb2, qb3);
    step8(acc, pa0, pa1, pb0, pb1, pb2, pb3);
    ld_ab6(xr0, xr1, wp0, wp1, wp2, wp3, k + 8 + hf2,
           pa0, pa1, pb0, pb1, pb2, pb3);
    step8(acc, qa0, qa1, qb0, qb1, qb2, qb3);
  }
  ld_ab6(xr0, xr1, wp0, wp1, wp2, wp3, H_ - 4 + hf2,
         qa0, qa1, qb0, qb1, qb2, qb3);
  step8(acc, pa0, pa1, pb0, pb1, pb2, pb3);
  step8(acc, qa0, qa1, qb0, qb1, qb2, qb3);

  #pragma unroll
  for (int t = 0; t < 2; ++t) {
    #pragma unroll
    for (int j = 0; j < 4; ++j) {
      const int col = n0 + j * 16 + l16;
      const float bov = bo[col];
      #pragma unroll
      for (int r = 0; r < 8; ++r) {
        const int sr = m0 + t * 16 + r + hf * 8;
        if (sr < S_)
          out[((size_t)b * S_ + sr) * H_ + col] = acc[t][j][r] + bov;
      }
    }
  }
}

// ---------------------------------------------------------------------------
extern "C" void kernel_launch(void* const* d_in, const int* in_sizes, int n_in,
                              void* d_out, int out_size, void* d_ws, size_t ws_size,
                              hipStream_t stream)
{
  (void)in_sizes; (void)n_in; (void)out_size; (void)ws_size;

  const float* X    = (const float*)d_in[0];
  const float* mask = (const float*)d_in[1];
  const float* Wq   = (const float*)d_in[2];
  const float* bq   = (const float*)d_in[3];
  const float* Wk   = (const float*)d_in[4];
  const float* bk   = (const float*)d_in[5];
  const float* Wv   = (const float*)d_in[6];
  const float* bv   = (const float*)d_in[7];
  const float* Wo   = (const float*)d_in[8];
  const float* bo   = (const float*)d_in[9];

  // Output tuple, flat: out (B,S,H) || weights (B,NH,S,S) || sm2 (B,NH,S)
  float* out     = (float*)d_out;
  float* weights = out + (size_t)B_ * S_ * H_;
  float* sm2     = weights + (size_t)B_ * NH_ * S_ * S_;

  // Workspace: Q,K head-major padded; V transposed; col0; ctx; W transposes
  const size_t qsz = (size_t)B_ * NH_ * SP_ * DH_;   // 7,864,320 floats each
  const size_t wsz = (size_t)H_ * H_;                // 589,824 floats each
  float* Q    = (float*)d_ws;
  float* K    = Q + qsz;
  float* Vt   = K + qsz;
  float* col0 = Vt + qsz;                            // B*NH*SP floats
  float* ctx  = col0 + (size_t)B_ * NH_ * SP_;       // B*SP*H floats
  float* Wqt  = ctx + (size_t)B_ * SP_ * H_;
  float* Wkt  = Wqt + wsz;
  float* Wvt  = Wkt + wsz;
  float* Wot  = Wvt + wsz;
  // total ≈ 136 MB of f32 workspace (fits in 192 MB L2)

  transpose_w<<<dim3(48, 48, 4), dim3(256), 0, stream>>>(
      Wq, Wk, Wv, Wo, Wqt, Wkt, Wvt, Wot);
  qkv_gemm<<<dim3(ST2_, NH_, B_ * 3), dim3(32), 0, stream>>>(
      X, Wqt, bq, Wkt, bk, Wvt, bv, Q, K, Vt);
  attn<<<dim3(ST_, NH_, B_), dim3(32), 0, stream>>>(
      Q, K, Vt, mask, weights, col0, ctx);
  sm2_softmax<<<dim3(B_ * NH_), dim3(32), 0, stream>>>(col0, sm2);
  out_gemm<<<dim3(ST2_, 12, B_), dim3(32), 0, stream>>>(ctx, Wot, bo, out);
}